// TinyTransformerBlock_89472758711002
// MI455X (gfx1250) — compile-verified
//
#include <hip/hip_runtime.h>
#include <hip/hip_bf16.h>

// ---------------------------------------------------------------------------
// TinyTransformerBlock on gfx1250 (MI455X), fp32 end-to-end using
// V_WMMA_F32_16X16X4_F32 for all four GEMMs, LDS-staged tiles for the two
// big conv1x1 GEMMs, K-split + LDS-reduce for attention scores.
//
// Shapes: B=8, C=256, HEADS=4, d=64, HW=4096.
// ---------------------------------------------------------------------------

typedef float v2f __attribute__((ext_vector_type(2)));
typedef float v8f __attribute__((ext_vector_type(8)));

#define BATCH   8
#define CDIM    256
#define QKVDIM  768
#define HEADS   4
#define HDIM    64
#define HW      4096
#define EPS     1e-5f

__device__ __forceinline__ v8f wmma4(v2f a, v2f b, v8f c) {
    // D = A(16x4, f32) * B(4x16, f32) + C(16x16, f32)
    return __builtin_amdgcn_wmma_f32_16x16x4_f32(
        /*neg_a=*/false, a, /*neg_b=*/false, b,
        /*c_mod=*/(short)0, c, /*reuse_a=*/false, /*reuse_b=*/false);
}

// A-fragment (16x4 tile at (m0,k0) of row-major M x lda matrix in GLOBAL mem):
// lane L: row = L&15, VGPR j holds K = k0 + 2*(L>>4) + j  -> one aligned float2
__device__ __forceinline__ v2f load_afrag(const float* __restrict__ A,
                                          int m0, int k0, int lda,
                                          int rw, int hl) {
    return *(const v2f*)&A[(size_t)(m0 + rw) * lda + k0 + 2 * hl];
}

// B-fragment (4x16 tile at (k0,n0) of row-major K x ldb matrix in GLOBAL mem)
__device__ __forceinline__ v2f load_bfrag(const float* __restrict__ B,
                                          int k0, int n0, int ldb,
                                          int rw, int hl) {
    v2f b;
    b.x = B[(size_t)(k0 + 2 * hl + 0) * ldb + n0 + rw];
    b.y = B[(size_t)(k0 + 2 * hl + 1) * ldb + n0 + rw];
    return b;
}

// ---------------------------------------------------------------------------
// LDS-staged conv1x1 GEMM:  O(m_rows x 4096) = W(m_rows x 256) * B(256 x 4096)
//                                               + bias (+ residual)
// Block = 256 threads (8 waves). Block tile = 128(M) x 64(N); waves form a
// 4(M) x 2(N) grid of 32x32 register tiles (2x2 WMMA blocking).
// K is consumed in slabs of 16 staged through LDS:
//   sW[128][20] (pad 16->20: conflict-free & float4-aligned)  = 10.0 KB
//   sX[16][68]  (pad 64->68: conflict-free & float4-aligned)  =  4.4 KB
// Next X slab is prefetched (global_prefetch_b8) while computing this one.
// Grid = (4096/64, m_rows/128, 8).
// ---------------------------------------------------------------------------
template <bool RESIDUAL>
__global__ void gemm128x64_lds(const float* __restrict__ W,
                               const float* __restrict__ B,
                               const float* __restrict__ bias,
                               const float* __restrict__ resid,
                               float* __restrict__ out,
                               int m_rows) {
    __shared__ float sW[128][20];
    __shared__ float sX[16][68];

    const int t    = threadIdx.x;
    const int lane = t & 31;
    const int wave = t >> 5;
    const int hl = lane >> 4, rw = lane & 15;

    const int mblk = blockIdx.y * 128;
    const int nblk = blockIdx.x * 64;
    const int b    = blockIdx.z;

    const float* Bb = B + (size_t)b * CDIM * HW;

    const int m0 = (wave & 3) * 32;   // register-tile origin within block tile
    const int n0 = (wave >> 2) * 32;

    // global->LDS staging indices
    const int wm = t >> 2;            // 0..63   (W row within slab half)
    const int wc = (t & 3) * 4;       // 0,4,8,12 (W col within slab)
    const int xr = t >> 4;            // 0..15   (X row within slab)
    const int xc = (t & 15) * 4;      // 0..60   (X col within tile)

    v8f c00 = {}, c01 = {}, c10 = {}, c11 = {};

    for (int k = 0; k < CDIM; k += 16) {
        __syncthreads();  // previous slab fully consumed
        *(float4*)&sW[wm][wc] =
            *(const float4*)&W[(size_t)(mblk + wm) * CDIM + k + wc];
        *(float4*)&sW[wm + 64][wc] =
            *(const float4*)&W[(size_t)(mblk + wm + 64) * CDIM + k + wc];
        *(float4*)&sX[xr][xc] =
            *(const float4*)&Bb[(size_t)(k + xr) * HW + nblk + xc];
        if (k + 16 < CDIM)  // prefetch next X slab into cache
            __builtin_prefetch(&Bb[(size_t)(k + 16 + xr) * HW + nblk + xc], 0, 0);
        __syncthreads();  // slab visible to all waves

#pragma unroll
        for (int kk = 0; kk < 16; kk += 4) {
            v2f a0 = *(const v2f*)&sW[m0 + rw][kk + 2 * hl];
            v2f a1 = *(const v2f*)&sW[m0 + 16 + rw][kk + 2 * hl];
            v2f b0, b1;
            b0.x = sX[kk + 2 * hl + 0][n0 + rw];
            b0.y = sX[kk + 2 * hl + 1][n0 + rw];
            b1.x = sX[kk + 2 * hl + 0][n0 + 16 + rw];
            b1.y = sX[kk + 2 * hl + 1][n0 + 16 + rw];
            c00 = wmma4(a0, b0, c00);
            c01 = wmma4(a0, b1, c01);
            c10 = wmma4(a1, b0, c10);
            c11 = wmma4(a1, b1, c11);
        }
    }

    float* O = out + (size_t)b * m_rows * HW;
    const float* R = resid + (size_t)b * m_rows * HW;
#pragma unroll
    for (int r = 0; r < 8; ++r) {
        const int mA = mblk + m0 + r + 8 * hl;
        const int mB = mA + 16;
        const float biasA = bias[mA], biasB = bias[mB];
        const size_t iA0 = (size_t)mA * HW + nblk + n0 + rw;
        const size_t iA1 = iA0 + 16;
        const size_t iB0 = (size_t)mB * HW + nblk + n0 + rw;
        const size_t iB1 = iB0 + 16;
        float vA0 = c00[r] + biasA, vA1 = c01[r] + biasA;
        float vB0 = c10[r] + biasB, vB1 = c11[r] + biasB;
        if (RESIDUAL) {
            vA0 += R[iA0]; vA1 += R[iA1];
            vB0 += R[iB0]; vB1 += R[iB1];
        }
        O[iA0] = vA0; O[iA1] = vA1;
        O[iB0] = vB0; O[iB1] = vB1;
    }
}

// ---------------------------------------------------------------------------
// Attention scores. Per (b,h): S(64x64) = scale * Q(64x4096) * K^T.
// Block = 256 threads (8 waves) per 16x16 output tile; each wave accumulates
// a K-chunk of 512, partial tiles are reduced through LDS.
// Grid = (4, 4, 32).
// ---------------------------------------------------------------------------
__global__ void attn_scores(const float* __restrict__ qkv,
                            float* __restrict__ attn) {
    __shared__ float sred[8][256];

    const int t    = threadIdx.x;
    const int lane = t & 31;
    const int wave = t >> 5;
    const int hl = lane >> 4, rw = lane & 15;
    const int bh = blockIdx.z;
    const int b = bh >> 2, h = bh & 3;

    const float* Q = qkv + ((size_t)b * QKVDIM + 0 * CDIM + h * HDIM) * HW;
    const float* K = qkv + ((size_t)b * QKVDIM + 1 * CDIM + h * HDIM) * HW;

    const int d0 = blockIdx.y * 16;
    const int f0 = blockIdx.x * 16;

    v8f c = {};
    const int e0 = wave * (HW / 8);
    for (int e = e0; e < e0 + HW / 8; e += 4) {
        v2f a  = load_afrag(Q, d0, e, HW, rw, hl);   // Q[d, e..e+3]
        v2f bt = load_afrag(K, f0, e, HW, rw, hl);   // K[f, e..e+3] == (K^T)[e, f]
        c = wmma4(a, bt, c);
    }
#pragma unroll
    for (int r = 0; r < 8; ++r) sred[wave][r * 32 + lane] = c[r];
    __syncthreads();

    // thread t <-> tile element (r = t>>5, lane2 = t&31)
    float s = 0.f;
#pragma unroll
    for (int w = 0; w < 8; ++w) s += sred[w][t];
    const int r2 = t >> 5, l2 = t & 31;
    const int row = d0 + r2 + 8 * (l2 >> 4);
    const int col = f0 + (l2 & 15);
    attn[(size_t)bh * HDIM * HDIM + (size_t)row * HDIM + col] = s * 0.125f;
}

// ---------------------------------------------------------------------------
// Softmax over last dim (64). One wave per row, 2 elems/lane; 2048 rows.
// ---------------------------------------------------------------------------
__global__ void softmax64(float* __restrict__ attn) {
    const int lane = threadIdx.x & 31;
    const int wave = threadIdx.x >> 5;
    const int row = blockIdx.x * 8 + wave;          // 0..2047
    float* p = attn + (size_t)row * HDIM;

    float v0 = p[lane], v1 = p[lane + 32];
    float m = fmaxf(v0, v1);
#pragma unroll
    for (int off = 16; off > 0; off >>= 1)
        m = fmaxf(m, __shfl_xor(m, off, 32));
    float e0 = __expf(v0 - m), e1 = __expf(v1 - m);
    float s = e0 + e1;
#pragma unroll
    for (int off = 16; off > 0; off >>= 1)
        s += __shfl_xor(s, off, 32);
    const float inv = 1.0f / s;
    p[lane]      = e0 * inv;
    p[lane + 32] = e1 * inv;
}

// ---------------------------------------------------------------------------
// Apply attention. Per (b,h): O(64x4096) = S(64x64) * V(64x4096).
// 32x32 per wave, block = 128 (4 waves). Grid = (4096/128, 64/32, 32).
// A (attn) is tiny (16 KB/head) and V rows stream once -> direct loads fine.
// ---------------------------------------------------------------------------
__global__ void attn_apply(const float* __restrict__ attn,
                           const float* __restrict__ qkv,
                           float* __restrict__ out) {
    const int lane = threadIdx.x & 31;
    const int wave = threadIdx.x >> 5;
    const int hl = lane >> 4, rw = lane & 15;
    const int bh = blockIdx.z;
    const int b = bh >> 2, h = bh & 3;

    const float* S = attn + (size_t)bh * HDIM * HDIM;
    const float* V = qkv + ((size_t)b * QKVDIM + 2 * CDIM + h * HDIM) * HW;
    float* O = out + ((size_t)b * CDIM + h * HDIM) * HW;

    const int m0 = blockIdx.y * 32;
    const int n0 = blockIdx.x * 128 + wave * 32;

    v8f c00 = {}, c01 = {}, c10 = {}, c11 = {};
#pragma unroll 4
    for (int k = 0; k < HDIM; k += 4) {
        v2f a0 = load_afrag(S, m0,      k, HDIM, rw, hl);
        v2f a1 = load_afrag(S, m0 + 16, k, HDIM, rw, hl);
        v2f b0 = load_bfrag(V, k, n0,      HW, rw, hl);
        v2f b1 = load_bfrag(V, k, n0 + 16, HW, rw, hl);
        c00 = wmma4(a0, b0, c00);
        c01 = wmma4(a0, b1, c01);
        c10 = wmma4(a1, b0, c10);
        c11 = wmma4(a1, b1, c11);
    }
#pragma unroll
    for (int r = 0; r < 8; ++r) {
        const int mA = m0 + r + 8 * hl;
        const int mB = mA + 16;
        O[(size_t)mA * HW + n0 +      rw] = c00[r];
        O[(size_t)mA * HW + n0 + 16 + rw] = c01[r];
        O[(size_t)mB * HW + n0 +      rw] = c10[r];
        O[(size_t)mB * HW + n0 + 16 + rw] = c11[r];
    }
}

// ---------------------------------------------------------------------------
// Per-channel mean / rstd over (batch, spatial) = 32768 values.
// One block per channel. Grid = 256, block = 256.
// ---------------------------------------------------------------------------
__global__ void channel_stats(const float* __restrict__ y,
                              float* __restrict__ mean,
                              float* __restrict__ rstd) {
    const int c = blockIdx.x;
    float s = 0.f, s2 = 0.f;
    for (int i = threadIdx.x; i < BATCH * HW; i += blockDim.x) {
        const int b = i >> 12;
        const int e = i & (HW - 1);
        const float v = y[((size_t)b * CDIM + c) * HW + e];
        s += v;
        s2 += v * v;
    }
    __shared__ float ls[256], ls2[256];
    ls[threadIdx.x] = s;
    ls2[threadIdx.x] = s2;
    __syncthreads();
    for (int off = 128; off > 0; off >>= 1) {
        if (threadIdx.x < off) {
            ls[threadIdx.x] += ls[threadIdx.x + off];
            ls2[threadIdx.x] += ls2[threadIdx.x + off];
        }
        __syncthreads();
    }
    if (threadIdx.x == 0) {
        const float inv_n = 1.0f / (float)(BATCH * HW);
        const float m = ls[0] * inv_n;
        const float var = ls2[0] * inv_n - m * m;
        mean[c] = m;
        rstd[c] = rsqrtf(var + EPS);
    }
}

// ---------------------------------------------------------------------------
// In-place normalize + affine.
// ---------------------------------------------------------------------------
__global__ void finalize_norm(float* __restrict__ y,
                              const float* __restrict__ mean,
                              const float* __restrict__ rstd,
                              const float* __restrict__ gamma,
                              const float* __restrict__ beta) {
    const size_t total = (size_t)BATCH * CDIM * HW;
    for (size_t i = (size_t)blockIdx.x * blockDim.x + threadIdx.x; i < total;
         i += (size_t)gridDim.x * blockDim.x) {
        const int c = (int)((i >> 12) & (CDIM - 1));
        y[i] = (y[i] - mean[c]) * rstd[c] * gamma[c] + beta[c];
    }
}

// ---------------------------------------------------------------------------
// Launch
// ---------------------------------------------------------------------------
extern "C" void kernel_launch(void* const* d_in, const int* in_sizes, int n_in,
                              void* d_out, int out_size, void* d_ws, size_t ws_size,
                              hipStream_t stream) {
    const float* x      = (const float*)d_in[0];
    const float* w_qkv  = (const float*)d_in[1];
    const float* b_qkv  = (const float*)d_in[2];
    const float* w_proj = (const float*)d_in[3];
    const float* b_proj = (const float*)d_in[4];
    const float* gamma  = (const float*)d_in[5];
    const float* beta   = (const float*)d_in[6];
    float* y = (float*)d_out;  // proj(out)+bias+x lives here, normalized in place

    float* ws   = (float*)d_ws;
    float* qkv  = ws;                                         // 8*768*4096
    float* attn = qkv + (size_t)BATCH * QKVDIM * HW;          // 8*4*64*64
    float* aout = attn + (size_t)BATCH * HEADS * HDIM * HDIM; // 8*256*4096
    float* mean = aout + (size_t)BATCH * CDIM * HW;           // 256
    float* rstd = mean + CDIM;                                // 256

    gemm128x64_lds<false><<<dim3(HW / 64, QKVDIM / 128, BATCH), 256, 0, stream>>>(
        w_qkv, x, b_qkv, x /*unused*/, qkv, QKVDIM);
    attn_scores  <<<dim3(4, 4, BATCH * HEADS),         256, 0, stream>>>(qkv, attn);
    softmax64    <<<dim3(256),                         256, 0, stream>>>(attn);
    attn_apply   <<<dim3(HW / 128, 2, BATCH * HEADS),  128, 0, stream>>>(attn, qkv, aout);
    gemm128x64_lds<true><<<dim3(HW / 64, CDIM / 128, BATCH), 256, 0, stream>>>(
        w_proj, aout, b_proj, x, y, CDIM);
    channel_stats<<<dim3(CDIM),                        256, 0, stream>>>(y, mean, rstd);
    finalize_norm<<<dim3(8192),                        256, 0, stream>>>(y, mean, rstd, gamma, beta);
}